// SAGE_1838246003329
// MI455X (gfx1250) — compile-verified
//
#include <hip/hip_runtime.h>
#include <math.h>

// GraphSAGE 3-layer forward for MI455X (gfx1250, wave32).
//
// Bottleneck analysis: 16.4 GFLOP of fp32 GEMM (trivial for WMMA) vs
// ~1.6 GB/layer of L2-resident gather + 205M fp32 atomics/layer for the
// scatter-mean. 51MB feature matrix fits in 192MB L2, so aggregation is
// on-chip. GEMMs use V_WMMA_F32_16X16X4_F32 (exact fp32 accumulate, matches
// the fp32 reference) with per-wave register-resident weight fragments.

#define N_NODES 100000
#define N_EDGES 1600000
#define BN_EPS  1e-5f

typedef float v2f __attribute__((ext_vector_type(2)));
typedef float v8f __attribute__((ext_vector_type(8)));

// ---------------------------------------------------------------- degrees
__global__ void count_deg_kernel(const int* __restrict__ dst,
                                 float* __restrict__ deg, int E) {
  int e = blockIdx.x * blockDim.x + threadIdx.x;
  if (e < E) atomicAdd(&deg[dst[e]], 1.0f);
}

// --------------------------------------------------- scatter-add (D=128)
// One thread handles 4 channels of one edge (float4 gather, 4 fp32 atomics).
// All traffic hits L2 (51MB << 192MB).
__global__ void scatter_add128_kernel(const float* __restrict__ x,
                                      const int* __restrict__ src,
                                      const int* __restrict__ dst,
                                      float* __restrict__ agg, int E) {
  int t = blockIdx.x * blockDim.x + threadIdx.x;
  if (t >= E * 32) return;
  int e  = t >> 5;
  int c4 = (t & 31) << 2;
  int s = src[e], d = dst[e];
  const float4 v = *(const float4*)(x + (size_t)s * 128 + c4);
  float* a = agg + (size_t)d * 128 + c4;
  atomicAdd(a + 0, v.x);
  atomicAdd(a + 1, v.y);
  atomicAdd(a + 2, v.z);
  atomicAdd(a + 3, v.w);
}

// ------------------------------------------------------------ SAGE GEMM
// out[r,c] = sum_k (agg[r,k]/max(deg[r],1)) * Wl[c,k] + sum_k x[r,k]*Wr[c,k]
// then epilogue: FUSE_BN ? relu((v+b-rm)*g*rsqrt(rv+eps)+be) : v+b
//
// One wave owns one 16-column tile (colTile = wid % NT) and keeps its B
// fragments for BOTH weight matrices in VGPRs (2x32 v2f = 128 VGPRs), then
// grid-strides over 16-row tiles. K = 128 (always), 64 WMMA per tile.
// N_NODES % 16 == 0, so no row-tile tail; EXEC stays all-ones for WMMA.
template <int NT, bool FUSE_BN>
__global__ __launch_bounds__(256) void sage_gemm_kernel(
    const float* __restrict__ Aagg,  // [N,128] un-normalized neighbor sums
    const float* __restrict__ Ax,    // [N,128] self features
    const float* __restrict__ deg,   // [N]
    const float* __restrict__ Wl,    // [NT*16,128]
    const float* __restrict__ Wr,    // [NT*16,128]
    const float* __restrict__ bias,  // [NT*16]
    const float* __restrict__ g, const float* __restrict__ be,
    const float* __restrict__ rm, const float* __restrict__ rv,
    float* __restrict__ out,         // [N, NT*16]
    int nRowTiles) {
  constexpr int K    = 128;
  constexpr int MOUT = NT * 16;

  const int wid   = (blockIdx.x * blockDim.x + threadIdx.x) >> 5;
  const int nw    = (gridDim.x * blockDim.x) >> 5;   // multiple of 8
  const int lane  = threadIdx.x & 31;
  const int n16   = lane & 15;          // A-row / B-col index within tile
  const int khalf = (lane >> 4) << 1;   // 0 for lanes 0-15, 2 for 16-31

  const int colTile = wid % NT;
  const int col     = colTile * 16 + n16;

  // Preload B fragments for this wave's column tile (register-resident).
  // ISA layout (4x16 f32 B): lane -> n = lane%16, k = 4*s + khalf + {0,1}.
  v2f bl[32], br[32];
  const float* wlrow = Wl + (size_t)col * K + khalf;
  const float* wrrow = Wr + (size_t)col * K + khalf;
#pragma unroll
  for (int s = 0; s < 32; ++s) {
    bl[s] = *(const v2f*)(wlrow + 4 * s);
    br[s] = *(const v2f*)(wrrow + 4 * s);
  }

  // Per-column epilogue constants (column is fixed per lane under C layout).
  float alpha, beta;
  if (FUSE_BN) {
    const float sc = g[col] * rsqrtf(rv[col] + BN_EPS);
    alpha = sc;
    beta  = be[col] + (bias[col] - rm[col]) * sc;
  } else {
    alpha = 1.0f;
    beta  = bias[col];
  }

  for (int r = wid / NT; r < nRowTiles; r += nw / NT) {
    const int rowA = r * 16 + n16;  // A layout: both lane-halves cover M=0..15
    const float rdeg = 1.0f / fmaxf(deg[rowA], 1.0f);  // fold mean into A
    const float* a1p = Aagg + (size_t)rowA * K + khalf;
    const float* a2p = Ax   + (size_t)rowA * K + khalf;

    v8f acc = {};
#pragma unroll
    for (int s = 0; s < 32; ++s) {
      v2f a1 = *(const v2f*)(a1p + 4 * s);
      a1.x *= rdeg;
      a1.y *= rdeg;
      acc = __builtin_amdgcn_wmma_f32_16x16x4_f32(
          false, a1, false, bl[s], (short)0, acc, false, false);
      const v2f a2 = *(const v2f*)(a2p + 4 * s);
      acc = __builtin_amdgcn_wmma_f32_16x16x4_f32(
          false, a2, false, br[s], (short)0, acc, false, false);
    }

    // C layout: VGPR v -> row = rbase + v (+8 for lanes 16-31), col = n16.
    const int rowBase = r * 16 + ((lane >> 4) << 3);
    float* op = out + (size_t)rowBase * MOUT + col;
#pragma unroll
    for (int v = 0; v < 8; ++v) {
      float val = acc[v] * alpha + beta;
      if (FUSE_BN) val = fmaxf(val, 0.0f);
      op[(size_t)v * MOUT] = val;
    }
  }
}

// ----------------------------------------------------- log_softmax (64)
// One wave per row, 2 values per lane, wave32 xor-shuffle reductions.
__global__ void log_softmax64_kernel(float* __restrict__ io, int N) {
  const int row  = (blockIdx.x * blockDim.x + threadIdx.x) >> 5;
  const int lane = threadIdx.x & 31;
  if (row >= N) return;
  float* p = io + (size_t)row * 64;
  const float v0 = p[lane], v1 = p[lane + 32];
  float m = fmaxf(v0, v1);
#pragma unroll
  for (int off = 16; off > 0; off >>= 1) m = fmaxf(m, __shfl_xor(m, off, 32));
  float s = expf(v0 - m) + expf(v1 - m);
#pragma unroll
  for (int off = 16; off > 0; off >>= 1) s += __shfl_xor(s, off, 32);
  const float lse = m + logf(s);
  p[lane]      = v0 - lse;
  p[lane + 32] = v1 - lse;
}

// ---------------------------------------------------------------- launch
extern "C" void kernel_launch(void* const* d_in, const int* in_sizes, int n_in,
                              void* d_out, int out_size, void* d_ws,
                              size_t ws_size, hipStream_t stream) {
  (void)in_sizes; (void)n_in; (void)out_size; (void)ws_size;

  const float* x   = (const float*)d_in[0];
  const int*   src = (const int*)d_in[1];
  const int*   dst = (const int*)d_in[2];
  const float* W1l = (const float*)d_in[3];
  const float* W1r = (const float*)d_in[4];
  const float* b1  = (const float*)d_in[5];
  const float* g1  = (const float*)d_in[6];
  const float* be1 = (const float*)d_in[7];
  const float* rm1 = (const float*)d_in[8];
  const float* rv1 = (const float*)d_in[9];
  const float* W2l = (const float*)d_in[10];
  const float* W2r = (const float*)d_in[11];
  const float* b2  = (const float*)d_in[12];
  const float* g2  = (const float*)d_in[13];
  const float* be2 = (const float*)d_in[14];
  const float* rm2 = (const float*)d_in[15];
  const float* rv2 = (const float*)d_in[16];
  const float* W3l = (const float*)d_in[17];
  const float* W3r = (const float*)d_in[18];
  const float* b3  = (const float*)d_in[19];

  // Workspace: deg[N] | agg[N*128] | h1[N*128] | h2[N*128]  (~154 MB)
  float* deg = (float*)d_ws;
  float* agg = deg + N_NODES;
  float* h1  = agg + (size_t)N_NODES * 128;
  float* h2  = h1 + (size_t)N_NODES * 128;
  float* outF = (float*)d_out;

  const int TPB = 256;
  const size_t aggBytes = (size_t)N_NODES * 128 * sizeof(float);
  const int nRT = N_NODES / 16;                       // 6250, exact
  const int GEMM_BLOCKS = 1024;                       // 8192 waves
  const int scatterGrid = (N_EDGES * 32 + TPB - 1) / TPB;

  // degrees (shared by all 3 layers)
  hipMemsetAsync(deg, 0, N_NODES * sizeof(float), stream);
  count_deg_kernel<<<(N_EDGES + TPB - 1) / TPB, TPB, 0, stream>>>(dst, deg, N_EDGES);

  // ---- layer 1: x -> h1
  hipMemsetAsync(agg, 0, aggBytes, stream);
  scatter_add128_kernel<<<scatterGrid, TPB, 0, stream>>>(x, src, dst, agg, N_EDGES);
  sage_gemm_kernel<8, true><<<GEMM_BLOCKS, TPB, 0, stream>>>(
      agg, x, deg, W1l, W1r, b1, g1, be1, rm1, rv1, h1, nRT);

  // ---- layer 2: h1 -> h2
  hipMemsetAsync(agg, 0, aggBytes, stream);
  scatter_add128_kernel<<<scatterGrid, TPB, 0, stream>>>(h1, src, dst, agg, N_EDGES);
  sage_gemm_kernel<8, true><<<GEMM_BLOCKS, TPB, 0, stream>>>(
      agg, h1, deg, W2l, W2r, b2, g2, be2, rm2, rv2, h2, nRT);

  // ---- layer 3: h2 -> d_out (N x 64), then in-place log_softmax
  hipMemsetAsync(agg, 0, aggBytes, stream);
  scatter_add128_kernel<<<scatterGrid, TPB, 0, stream>>>(h2, src, dst, agg, N_EDGES);
  sage_gemm_kernel<4, false><<<GEMM_BLOCKS, TPB, 0, stream>>>(
      agg, h2, deg, W3l, W3r, b3, nullptr, nullptr, nullptr, nullptr, outF, nRT);
  log_softmax64_kernel<<<(N_NODES * 32 + TPB - 1) / TPB, TPB, 0, stream>>>(outF, N_NODES);
}